// LearnedRouter_38139309589085
// MI455X (gfx1250) — compile-verified
//
#include <hip/hip_runtime.h>
#include <math.h>

typedef __attribute__((ext_vector_type(2))) float v2f;
typedef __attribute__((ext_vector_type(8))) float v8f;

#define N_ELEM   16777216
#define K_TOP    64
#define T_STEPS  20
#define EPS_F    0.1f
#define INV_EPS  10.0f
#define LOG_K    4.158883083f   /* ln(64) */
#define NBLK     2048
#define NTHR     256
#define NBINS    8192           /* top 13 bits of float key */
#define BIN_SHIFT 19
#define CAND_MAX 32768

/* workspace layout (bytes) */
#define OFF_STATE   0       /* float[16]: [0]=sigma [1]=tau [2]=a [3]=M */
#define OFF_MAXKEY  64      /* u32 */
#define OFF_CANDCNT 68      /* u32 */
#define OFF_THR     72      /* u32 */
#define OFF_PART    256     /* float[NBLK] */
#define OFF_HIST    8704    /* u32[NBINS]  */
#define OFF_CAND    41472   /* u64[CAND_MAX] */

__device__ __forceinline__ unsigned enc_f32(float f) {
    unsigned u = __float_as_uint(f);
    return u ^ ((u >> 31) ? 0xFFFFFFFFu : 0x80000000u);
}
__device__ __forceinline__ float dec_f32(unsigned k) {
    k ^= ((k >> 31) ? 0x80000000u : 0xFFFFFFFFu);
    return __uint_as_float(k);
}

/* wave32 sum of (acc0+acc1) over all 32 lanes via V_WMMA_F32_16X16X4_F32:
   A = 16x4 f32 (2 VGPRs/lane = 64 wave partials), B = ones(4x16).
   D rows hold 4-way sums; sum the 8 D regs (rows 0-7 / 8-15), then xor-16. */
__device__ __forceinline__ float wave_sum_wmma(float acc0, float acc1) {
    v2f a; a[0] = acc0; a[1] = acc1;
    v2f b; b[0] = 1.0f; b[1] = 1.0f;
    v8f c = {};
    v8f d = __builtin_amdgcn_wmma_f32_16x16x4_f32(
        false, a, false, b, (short)0, c, false, false);
    float r = ((d[0] + d[1]) + (d[2] + d[3])) + ((d[4] + d[5]) + (d[6] + d[7]));
    r += __shfl_xor(r, 16, 32);
    return r;
}

__global__ void init_kernel(unsigned* hist, unsigned* maxkey, unsigned* candcnt) {
    int i = blockIdx.x * blockDim.x + threadIdx.x;
    if (i < NBINS) hist[i] = 0u;
    if (i == 0) { *maxkey = 0u; *candcnt = 0u; }
}

__global__ void max_kernel(const float4* __restrict__ x4,
                           const float4* __restrict__ w4,
                           unsigned* maxkey) {
    int tid = blockIdx.x * blockDim.x + threadIdx.x;
    int stride = gridDim.x * blockDim.x;
    float m = -INFINITY;
    for (int i = tid; i < N_ELEM / 4; i += stride) {
        float4 xv = x4[i], wv = w4[i];
        float s0 = xv.x * wv.x, s1 = xv.y * wv.y;
        float s2 = xv.z * wv.z, s3 = xv.w * wv.w;
        m = fmaxf(m, fmaxf(fmaxf(s0, s1), fmaxf(s2, s3)));
    }
    for (int off = 16; off > 0; off >>= 1)
        m = fmaxf(m, __shfl_xor(m, off, 32));
    __shared__ float lds[NTHR / 32];
    int lane = threadIdx.x & 31, w = threadIdx.x >> 5;
    if (lane == 0) lds[w] = m;
    __syncthreads();
    if (threadIdx.x == 0) {
        float t = lds[0];
        for (int i = 1; i < NTHR / 32; ++i) t = fmaxf(t, lds[i]);
        atomicMax(maxkey, enc_f32(t));
    }
}

__global__ void seed_kernel(const unsigned* maxkey, float* state) {
    float M = dec_f32(*maxkey);
    state[3] = M;
    state[0] = M;          /* sigma */
    state[1] = INFINITY;   /* tau   */
    state[2] = 0.0f;       /* a     */
}

__global__ void sum_kernel(const float4* __restrict__ x4,
                           const float4* __restrict__ w4,
                           const float* __restrict__ state,
                           float* __restrict__ partials) {
    const float sigma = state[0];
    const float tau   = state[1];
    int tid = blockIdx.x * blockDim.x + threadIdx.x;
    int stride = gridDim.x * blockDim.x;
    float acc0 = 0.0f, acc1 = 0.0f;
    for (int i = tid; i < N_ELEM / 4; i += stride) {
        float4 xv = x4[i], wv = w4[i];
        float s0 = xv.x * wv.x, s1 = xv.y * wv.y;
        float s2 = xv.z * wv.z, s3 = xv.w * wv.w;
        acc0 += __expf((fminf(s0, tau) - sigma) * INV_EPS);
        acc1 += __expf((fminf(s1, tau) - sigma) * INV_EPS);
        acc0 += __expf((fminf(s2, tau) - sigma) * INV_EPS);
        acc1 += __expf((fminf(s3, tau) - sigma) * INV_EPS);
    }
    float r = wave_sum_wmma(acc0, acc1);   /* full wave active: EXEC all-1s */
    __shared__ float lds[NTHR / 32];
    int lane = threadIdx.x & 31, w = threadIdx.x >> 5;
    if (lane == 0) lds[w] = r;
    __syncthreads();
    if (threadIdx.x == 0) {
        float t = 0.0f;
        for (int i = 0; i < NTHR / 32; ++i) t += lds[i];  /* fixed order */
        partials[blockIdx.x] = t;
    }
}

__global__ void combine_kernel(const float* __restrict__ partials, float* state) {
    __shared__ float buf[NTHR];
    float t = 0.0f;
    for (int i = threadIdx.x; i < NBLK; i += NTHR) t += partials[i]; /* fixed order */
    buf[threadIdx.x] = t;
    __syncthreads();
    for (int off = NTHR / 2; off > 0; off >>= 1) {
        if (threadIdx.x < off) buf[threadIdx.x] += buf[threadIdx.x + off];
        __syncthreads();
    }
    if (threadIdx.x == 0) {
        float S = buf[0];
        float sigma = state[0];
        float lse = sigma * INV_EPS + __logf(S);
        float a = EPS_F * (LOG_K - lse);
        state[2] = a;
        float tau = -a;
        state[1] = tau;
        state[0] = fminf(state[3], tau);   /* next shift, keeps exp args <= 0 */
    }
}

__device__ __forceinline__ unsigned lam_key(float s, float a) {
    float e = (s + a) * INV_EPS;
    float lam = (e < 0.0f) ? __expf(e) : 1.0f;   /* lam in (0,1], monotone in s */
    return __float_as_uint(lam);                 /* positive float: bits monotone */
}

__global__ void hist_kernel(const float4* __restrict__ x4,
                            const float4* __restrict__ w4,
                            const float* __restrict__ state,
                            unsigned* __restrict__ hist,
                            float4* __restrict__ m4) {
    __shared__ unsigned lhist[NBINS];  /* 32 KB */
    for (int i = threadIdx.x; i < NBINS; i += NTHR) lhist[i] = 0u;
    __syncthreads();
    const float a = state[2];
    const unsigned floor_key = 0x30000000u;  /* ~4.6e-10: skip negligible lam */
    const float4 z4 = make_float4(0.f, 0.f, 0.f, 0.f);
    int tid = blockIdx.x * blockDim.x + threadIdx.x;
    int stride = gridDim.x * blockDim.x;
    for (int i = tid; i < N_ELEM / 4; i += stride) {
        float4 xv = x4[i], wv = w4[i];
        unsigned k0 = lam_key(xv.x * wv.x, a), k1 = lam_key(xv.y * wv.y, a);
        unsigned k2 = lam_key(xv.z * wv.z, a), k3 = lam_key(xv.w * wv.w, a);
        m4[i] = z4;  /* zero-fill output m */
        if (k0 >= floor_key) atomicAdd(&lhist[k0 >> BIN_SHIFT], 1u);
        if (k1 >= floor_key) atomicAdd(&lhist[k1 >> BIN_SHIFT], 1u);
        if (k2 >= floor_key) atomicAdd(&lhist[k2 >> BIN_SHIFT], 1u);
        if (k3 >= floor_key) atomicAdd(&lhist[k3 >> BIN_SHIFT], 1u);
    }
    __syncthreads();
    for (int i = threadIdx.x; i < NBINS; i += NTHR)
        if (lhist[i]) atomicAdd(&hist[i], lhist[i]);
}

__global__ void thresh_kernel(const unsigned* __restrict__ hist, unsigned* thr) {
    unsigned cum = 0, tk = 0;
    for (int i = NBINS - 1; i >= 0; --i) {
        cum += hist[i];
        if (cum >= K_TOP) { tk = ((unsigned)i) << BIN_SHIFT; break; }
    }
    *thr = tk;
}

__global__ void cand_kernel(const float4* __restrict__ x4,
                            const float4* __restrict__ w4,
                            const float* __restrict__ state,
                            const unsigned* __restrict__ thr,
                            unsigned* __restrict__ cnt,
                            unsigned long long* __restrict__ cands) {
    const float a = state[2];
    const unsigned tk = *thr;
    int tid = blockIdx.x * blockDim.x + threadIdx.x;
    int stride = gridDim.x * blockDim.x;
    for (int i = tid; i < N_ELEM / 4; i += stride) {
        float4 xv = x4[i], wv = w4[i];
        float s[4] = { xv.x * wv.x, xv.y * wv.y, xv.z * wv.z, xv.w * wv.w };
        #pragma unroll
        for (int j = 0; j < 4; ++j) {
            unsigned key = lam_key(s[j], a);
            if (key >= tk) {
                unsigned pos = atomicAdd(cnt, 1u);
                if (pos < CAND_MAX) {
                    unsigned idx = (unsigned)(4 * i + j);
                    cands[pos] = ((unsigned long long)key << 32) |
                                 (unsigned long long)(0xFFFFFFFFu - idx);
                }
            }
        }
    }
}

__global__ void topk_kernel(const unsigned* __restrict__ cnt,
                            unsigned long long* __restrict__ cands,
                            float* __restrict__ out_m,
                            float* __restrict__ out_idx) {
    __shared__ unsigned long long sb[1024];
    __shared__ int si[1024];
    int n = (int)*cnt;
    if (n > CAND_MAX) n = CAND_MAX;
    for (int r = 0; r < K_TOP; ++r) {
        unsigned long long best = 0ull; int bi = -1;
        for (int i = threadIdx.x; i < n; i += 1024)
            if (cands[i] > best) { best = cands[i]; bi = i; }
        sb[threadIdx.x] = best; si[threadIdx.x] = bi;
        __syncthreads();
        for (int off = 512; off > 0; off >>= 1) {
            if (threadIdx.x < off && sb[threadIdx.x + off] > sb[threadIdx.x]) {
                sb[threadIdx.x] = sb[threadIdx.x + off];
                si[threadIdx.x] = si[threadIdx.x + off];
            }
            __syncthreads();
        }
        if (threadIdx.x == 0) {
            unsigned long long bk = sb[0];
            if (bk != 0ull) {
                unsigned idx = 0xFFFFFFFFu - (unsigned)(bk & 0xFFFFFFFFull);
                float lam = __uint_as_float((unsigned)(bk >> 32));
                out_idx[r] = (float)idx;   /* <= 2^24: exact in f32 */
                out_m[idx] = lam;
                cands[si[0]] = 0ull;
            } else {
                out_idx[r] = 0.0f;
            }
        }
        __syncthreads();
    }
}

extern "C" void kernel_launch(void* const* d_in, const int* in_sizes, int n_in,
                              void* d_out, int out_size, void* d_ws, size_t ws_size,
                              hipStream_t stream) {
    (void)in_sizes; (void)n_in; (void)out_size; (void)ws_size;
    const float4* x4 = (const float4*)d_in[0];
    const float4* w4 = (const float4*)d_in[1];
    float* out = (float*)d_out;
    char* ws = (char*)d_ws;

    float*    state   = (float*)(ws + OFF_STATE);
    unsigned* maxkey  = (unsigned*)(ws + OFF_MAXKEY);
    unsigned* candcnt = (unsigned*)(ws + OFF_CANDCNT);
    unsigned* thr     = (unsigned*)(ws + OFF_THR);
    float*    part    = (float*)(ws + OFF_PART);
    unsigned* hist    = (unsigned*)(ws + OFF_HIST);
    unsigned long long* cands = (unsigned long long*)(ws + OFF_CAND);

    init_kernel<<<(NBINS + NTHR - 1) / NTHR, NTHR, 0, stream>>>(hist, maxkey, candcnt);
    max_kernel<<<NBLK, NTHR, 0, stream>>>(x4, w4, maxkey);
    seed_kernel<<<1, 1, 0, stream>>>(maxkey, state);
    for (int t = 0; t < T_STEPS; ++t) {
        sum_kernel<<<NBLK, NTHR, 0, stream>>>(x4, w4, state, part);
        combine_kernel<<<1, NTHR, 0, stream>>>(part, state);
    }
    hist_kernel<<<NBLK, NTHR, 0, stream>>>(x4, w4, state, hist, (float4*)out);
    thresh_kernel<<<1, 1, 0, stream>>>(hist, thr);
    cand_kernel<<<NBLK, NTHR, 0, stream>>>(x4, w4, state, thr, candcnt, cands);
    topk_kernel<<<1, 1024, 0, stream>>>(candcnt, cands, out, out + N_ELEM);
}